// Voxelizer_14087492731380
// MI455X (gfx1250) — compile-verified
//
#include <hip/hip_runtime.h>
#include <hip/hip_bf16.h>

typedef float v2f __attribute__((ext_vector_type(2)));
typedef float v8f __attribute__((ext_vector_type(8)));

#define GDIM   128          // D == H == W == 128
#define HALF   64.0f
#define INVHALF 0.015625f   // 1/64, exact power of two
#define KWIN   12
#define GCUT   3.0f
// -0.5 * log2(e): folds exp(-0.5*m) into exp2(m') with pre-scaled coefficients
#define NEGHALF_LOG2E (-0.72134752044448170367996234050095f)

// One wave (32 lanes) per Gaussian. Per x-slice, one V_WMMA_F32_16X16X4_F32
// evaluates the full 16x16 (z,y) plane of exp2 arguments:
//   arg(z,y|x) = A0(z,x)*1 + A1(z,x)*y + c11'*y^2 + log2(density)
// where coefficients are pre-scaled by -0.5*log2(e) and log2(density) rides
// in the WMMA C accumulator. Weight = raw v_exp_f32 per voxel (denormal-range
// weights < 2^-126 flush to 0; they are numerically irrelevant here).
__global__ __launch_bounds__(256) void voxelize_wmma_kernel(
    const float* __restrict__ pos,   // (N,3) dims (z=D, y=H, x=W)
    const float* __restrict__ scl,   // (N,3)
    const float* __restrict__ rot,   // (N,4) (w,x,y,z)
    const float* __restrict__ dens,  // (N,)
    float* __restrict__ out,         // (128^3,)
    int n)
{
    const int lane = threadIdx.x & 31;
    const int wave = threadIdx.x >> 5;
    const int g    = blockIdx.x * 8 + wave;
    if (g >= n) return;                       // wave-uniform branch: EXEC stays full

    // ---- per-Gaussian (wave-uniform) setup ----------------------------------
    const float p0 = pos[3*g+0], p1 = pos[3*g+1], p2 = pos[3*g+2];
    const float s0 = scl[3*g+0], s1 = scl[3*g+1], s2 = scl[3*g+2];
    float qw = rot[4*g+0], qx = rot[4*g+1], qy = rot[4*g+2], qz = rot[4*g+3];
    const float den = dens[g];
    const float l2den = __log2f(den);         // -inf for den==0 -> exp2 -> 0

    const float qn = rsqrtf(qw*qw + qx*qx + qy*qy + qz*qz);
    qw *= qn; qx *= qn; qy *= qn; qz *= qn;

    const float r00 = 1.0f - 2.0f*(qy*qy + qz*qz);
    const float r01 = 2.0f*(qx*qy - qw*qz);
    const float r02 = 2.0f*(qx*qz + qw*qy);
    const float r10 = 2.0f*(qx*qy + qw*qz);
    const float r11 = 1.0f - 2.0f*(qx*qx + qz*qz);
    const float r12 = 2.0f*(qy*qz - qw*qx);
    const float r20 = 2.0f*(qx*qz - qw*qy);
    const float r21 = 2.0f*(qy*qz + qw*qx);
    const float r22 = 1.0f - 2.0f*(qx*qx + qy*qy);

    const float i0 = 1.0f / (s0*s0 + 1e-8f);
    const float i1 = 1.0f / (s1*s1 + 1e-8f);
    const float i2 = 1.0f / (s2*s2 + 1e-8f);

    // covariance-inverse coefficients, pre-scaled by -0.5*log2(e)
    const float sc  = NEGHALF_LOG2E;
    const float c00 = (r00*r00*i0 + r01*r01*i1 + r02*r02*i2) * sc;
    const float c01 = (r00*r10*i0 + r01*r11*i1 + r02*r12*i2) * sc;
    const float c02 = (r00*r20*i0 + r01*r21*i1 + r02*r22*i2) * sc;
    const float c11 = (r10*r10*i0 + r11*r11*i1 + r12*r12*i2) * sc;
    const float c12 = (r10*r20*i0 + r11*r21*i1 + r12*r22*i2) * sc;
    const float c22 = (r20*r20*i0 + r21*r21*i1 + r22*r22*i2) * sc;

    const float pv0 = (p0 + 1.0f)*HALF - 0.5f;
    const float pv1 = (p1 + 1.0f)*HALF - 0.5f;
    const float pv2 = (p2 + 1.0f)*HALF - 0.5f;
    const float radius = fmaxf(s0, fmaxf(s1, s2)) * HALF * GCUT;

    const int min0 = min(max((int)floorf(pv0 - radius), 0), GDIM-1);
    const int min1 = min(max((int)floorf(pv1 - radius), 0), GDIM-1);
    const int min2 = min(max((int)floorf(pv2 - radius), 0), GDIM-1);
    const int max0 = min(max((int)ceilf(pv0 + radius), 0), GDIM-1) + 1;
    const int max1 = min(max((int)ceilf(pv1 + radius), 0), GDIM-1) + 1;
    const int max2 = min(max((int)ceilf(pv2 + radius), 0), GDIM-1) + 1;
    const int cnt0 = min(max0 - min0, KWIN);
    const int cnt1 = min(max1 - min1, KWIN);
    const int cnt2 = min(max2 - min2, KWIN);

    // ---- per-lane WMMA operand prep ----------------------------------------
    const int sub = lane & 15;
    const int hi  = lane >> 4;      // 0: lanes 0-15 (K=0,1), 1: lanes 16-31 (K=2,3)

    const float dzL = ((float)(min0 + sub) + 0.5f) * INVHALF - 1.0f - p0;
    const float dyL = ((float)(min1 + sub) + 0.5f) * INVHALF - 1.0f - p1;

    // B (4x16): rows [1, y, y^2, 0]; lanes 0-15 hold K=0,1; lanes 16-31 hold K=2,3
    v2f b;
    b.x = hi ? (dyL * dyL) : 1.0f;
    b.y = hi ? 0.0f        : dyL;

    // per-lane loop-invariant pieces of A0(z,x), A1(z,x)
    const float pA0 = c00 * dzL * dzL;   // c00' z^2
    const float pA1 = 2.0f * c02 * dzL;  // 2 c02' z  (coeff of x)
    const float pA2 = 2.0f * c01 * dzL;  // 2 c01' z
    const float k12 = 2.0f * c12;

    // C accumulator = log2(density) broadcast: density multiply rides the WMMA
    const v8f cinit = { l2den, l2den, l2den, l2den, l2den, l2den, l2den, l2den };

    // ---- x-slice loop (wave-uniform trip count) -----------------------------
    for (int xs = 0; xs < cnt2; ++xs) {
        const float dxv = ((float)(min2 + xs) + 0.5f) * INVHALF - 1.0f - p2;

        // A (16x4): rows [A0, A1, c11', 0]; lanes 0-15 hold K=0,1; lanes 16-31 K=2,3
        const float a0 = fmaf(fmaf(c22, dxv, pA1), dxv, pA0); // c22'x^2 + 2c02'zx + c00'z^2
        const float a1 = fmaf(k12, dxv, pA2);                 // 2c01'z + 2c12'x
        v2f a;
        a.x = hi ? c11  : a0;
        a.y = hi ? 0.0f : a1;

        // D(16x16) = A(16x4) x B(4x16) + log2(den) -> exp2 arguments for all (z,y)
        v8f acc = __builtin_amdgcn_wmma_f32_16x16x4_f32(
            /*neg_a=*/false, a, /*neg_b=*/false, b,
            /*c_mod=*/(short)0, cinit, /*reuse_a=*/false, /*reuse_b=*/false);

        // lane holds D rows z = r + 8*hi, column y = sub
        const int ybase = ((min1 + sub) * GDIM) + (min2 + xs);
        const bool yok = (sub < cnt1);
        #pragma unroll
        for (int r = 0; r < 8; ++r) {
            const int zo = r + 8 * hi;
            if (yok && zo < cnt0) {
                // raw v_exp_f32 (flush-to-zero below 2^-126 is fine here)
                const float wgt = __builtin_amdgcn_exp2f(acc[r]);
                unsafeAtomicAdd(&out[(min0 + zo) * (GDIM * GDIM) + ybase], wgt);
            }
        }
    }
}

extern "C" void kernel_launch(void* const* d_in, const int* in_sizes, int n_in,
                              void* d_out, int out_size, void* d_ws, size_t ws_size,
                              hipStream_t stream) {
    const float* pos  = (const float*)d_in[0];
    const float* scl  = (const float*)d_in[1];
    const float* rot  = (const float*)d_in[2];
    const float* dens = (const float*)d_in[3];
    float* out = (float*)d_out;
    const int n = in_sizes[0] / 3;

    // harness poisons d_out with 0xAA; accumulation needs zeros (capture-legal)
    hipMemsetAsync(out, 0, (size_t)out_size * sizeof(float), stream);

    const int blocks = (n + 7) / 8;   // 8 waves (8 Gaussians) per 256-thread block
    voxelize_wmma_kernel<<<blocks, 256, 0, stream>>>(pos, scl, rot, dens, out, n);
}